// NequIP_64759516889477
// MI455X (gfx1250) — compile-verified
//
#include <hip/hip_runtime.h>
#include <math.h>

// ---------------------------------------------------------------------------
// NequIP forward for MI455X (gfx1250, wave32).
// Strategy:
//  * all intermediates live in d_ws; agg buffer (29MB) is L2-resident on MI455X
//  * edge TP -> atomicAdd scatter (no per-edge materialization)
//  * structurally-dead TP paths (l3>=3; layer3 l3>=1) are pruned
//  * node linears use V_WMMA_F32_16X16X4_F32 (native f32 matrix pipe)
//  * CG tables + act-normalization constants generated on device each launch
// ---------------------------------------------------------------------------

typedef __attribute__((ext_vector_type(2))) float v2f;
typedef __attribute__((ext_vector_type(8))) float v8f;

#define DENOM_INV (1.0f/1.5f)

// node feature layout (192 floats/node), order = TARGET_MID order
#define X0P 0    // 32
#define X0M 32   // 32
#define X1P 64   // 8 x 3 (u-major)
#define X1M 88   // 8 x 3
#define X2P 112  // 8 x 5
#define X2M 152  // 8 x 5

// agg layout (1456 floats/node): blocks ordered exactly as linear_apply's xin
// A0P@0(48), A0M@48(48), A1P@96(80x3), A1M@336(80x3), A2P@576(88x5), A2M@1016(88x5)

// CG table float offsets inside ws
#define CG112 0
#define CG213 45
#define CG110 150
#define CG111 159
#define CG121 186
#define CG122 231
#define CG132 306
#define CG211 411
#define CG212 456
#define CG220 531
#define CG221 556
#define CG222 631
#define CG231 756
#define CG232 861
// total 1036 floats

__constant__ int CG_TRIPLES[14][4] = {
  {1,1,2,CG112},{2,1,3,CG213},{1,1,0,CG110},{1,1,1,CG111},{1,2,1,CG121},
  {1,2,2,CG122},{1,3,2,CG132},{2,1,1,CG211},{2,1,2,CG212},{2,2,0,CG220},
  {2,2,1,CG221},{2,2,2,CG222},{2,3,1,CG231},{2,3,2,CG232}};

// mul=8 TP paths: {srcNodeOff, l1, yOff, l2, cgOff, l3, dstAggOff, ubase}
// first 4 entries are the l3=0 paths (the only ones layer 3 needs)
__constant__ int PATHS[26][8] = {
  {X1M,1,0,1,CG110,0,   0,32}, {X2P,2,3,2,CG220,0,   0,40},
  {X1P,1,0,1,CG110,0,  48,32}, {X2M,2,3,2,CG220,0,  48,40},
  {X1P,1,3,2,CG121,1,  96,40}, {X1M,1,0,1,CG111,1,  96,48},
  {X2P,2,3,2,CG221,1,  96,56}, {X2M,2,0,1,CG211,1,  96,64},
  {X2M,2,8,3,CG231,1,  96,72},
  {X1P,1,0,1,CG111,1, 336,40}, {X1M,1,3,2,CG121,1, 336,48},
  {X2P,2,0,1,CG211,1, 336,56}, {X2P,2,8,3,CG231,1, 336,64},
  {X2M,2,3,2,CG221,1, 336,72},
  {X1P,1,3,2,CG122,2, 576,40}, {X1M,1,0,1,CG112,2, 576,48},
  {X1M,1,8,3,CG132,2, 576,56}, {X2P,2,3,2,CG222,2, 576,64},
  {X2M,2,0,1,CG212,2, 576,72}, {X2M,2,8,3,CG232,2, 576,80},
  {X1P,1,0,1,CG112,2,1016,40}, {X1P,1,8,3,CG132,2,1016,48},
  {X1M,1,3,2,CG122,2,1016,56}, {X2P,2,0,1,CG212,2,1016,64},
  {X2P,2,8,3,CG232,2,1016,72}, {X2M,2,3,2,CG222,2,1016,80}};

// mul=32 paths (scalar x0 times y, cg = identity): {srcOff, yOff, nk, dstAggOff}, ubase=8
__constant__ int MUL32P[4][4] = { {X0M,0,3,96},{X0P,0,3,336},{X0P,3,5,576},{X0M,3,5,1016} };

// passthrough copies: {srcNodeOff, dstAggOff, len}; layer3 uses first 2
__constant__ int PASSB[6][3] = { {X0P,0,32},{X0M,48,32},{X1P,96,24},{X1M,336,24},{X2P,576,40},{X2M,1016,40} };

// ------------------------------ init kernels -------------------------------

__device__ double dfact(int n){ double r=1.0; for(int i=2;i<=n;i++) r*=(double)i; return r; }

__device__ double su2cg(int j1,int m1,int j2,int m2,int j3,int m3){
  if(m3 != m1+m2) return 0.0;
  int vmin = -j1 + j2 + m3;
  if(-j1+m1 > vmin) vmin = -j1+m1;
  if(vmin < 0) vmin = 0;
  int vmax = j2 + j3 + m1;
  if(j3-j1+j2 < vmax) vmax = j3-j1+j2;
  if(j3+m3   < vmax) vmax = j3+m3;
  double C = sqrt((2.0*j3+1.0)*dfact(j3+j1-j2)*dfact(j3-j1+j2)*dfact(j1+j2-j3)/dfact(j1+j2+j3+1))
           * sqrt(dfact(j3+m3)*dfact(j3-m3)/(dfact(j1+m1)*dfact(j1-m1)*dfact(j2+m2)*dfact(j2-m2)));
  double S = 0.0;
  for(int v=vmin; v<=vmax; v++){
    double sg = ((v+j2+m2)&1)? -1.0 : 1.0;
    S += sg * dfact(j2+j3+m1-v)*dfact(j1-m1+v)
       / (dfact(v)*dfact(j3-j1+j2-v)*dfact(j3+m3-v)*dfact(v+j1-j2-m3));
  }
  return C*S;
}

// complex element of change-of-basis q_l[r][c] (reference _q), times (-1j)^l
__device__ void qmat(int l,int r,int c,double* qr,double* qi){
  int m = r - l;
  double ar=0.0, ai=0.0;
  const double s2 = 0.70710678118654752440;
  if(m<0){
    if(c == l-m) ar = s2;      // col l+|m|
    if(c == l+m) ai = -s2;     // col l-|m|
  } else if(m==0){
    if(c==l) ar = 1.0;
  } else {
    double sg = (m&1)? -1.0:1.0;
    if(c == l+m) ar = sg*s2;
    if(c == l-m) ai = sg*s2;
  }
  double pr,pi;
  switch(l&3){ case 0: pr=1; pi=0; break; case 1: pr=0; pi=-1; break;
               case 2: pr=-1; pi=0; break; default: pr=0; pi=1; break; }
  *qr = ar*pr - ai*pi;
  *qi = ar*pi + ai*pr;
}

__global__ void cg_init_kernel(float* __restrict__ cg){
  int tid = threadIdx.x + blockIdx.x*blockDim.x;
  int nth = blockDim.x*gridDim.x;
  for(int t=0;t<14;t++){
    int l1=CG_TRIPLES[t][0], l2=CG_TRIPLES[t][1], l3=CG_TRIPLES[t][2], off=CG_TRIPLES[t][3];
    int n1=2*l1+1, n2=2*l2+1, n3=2*l3+1, sz=n1*n2*n3;
    for(int idx=tid; idx<sz; idx+=nth){
      int j = idx/(n2*n3); int rem = idx%(n2*n3); int l = rem/n3; int n = rem%n3;
      double cr = 0.0;
      for(int i=0;i<n1;i++){
        double q1r,q1i; qmat(l1,i,j,&q1r,&q1i);
        if(q1r==0.0 && q1i==0.0) continue;
        for(int k=0;k<n2;k++){
          double q2r,q2i; qmat(l2,k,l,&q2r,&q2i);
          if(q2r==0.0 && q2i==0.0) continue;
          for(int m=0;m<n3;m++){
            double q3r,q3i; qmat(l3,m,n,&q3r,&q3i);
            if(q3r==0.0 && q3i==0.0) continue;
            double s = su2cg(l1,i-l1,l2,k-l2,l3,m-l3);
            if(s==0.0) continue;
            double ar = q1r*q2r - q1i*q2i;
            double ai = q1r*q2i + q1i*q2r;
            cr += (ar*q3r + ai*q3i)*s;  // real part of q1*q2*conj(q3)
          }
        }
      }
      cg[off+idx] = (float)cr;
    }
  }
}

// reproduce _norm_const() trapezoid integral; cst[0]=1/C_GELU, cst[1]=1/C_TANH
__global__ void norm_init_kernel(float* __restrict__ cst){
  __shared__ double sg[256], st[256];
  double ag=0.0, at=0.0;
  const double dz = 20.0/100000.0;
  for(int i=threadIdx.x;i<100001;i+=256){
    double z = -10.0 + dz*(double)i;
    double w = exp(-0.5*z*z)/sqrt(2.0*3.14159265358979323846);
    double g = 0.5*z*(1.0+tanh(0.79788456080286535588*(z+0.044715*z*z*z)));
    double t = tanh(z);
    double hw = (i==0 || i==100000) ? 0.5 : 1.0;
    ag += hw*g*g*w; at += hw*t*t*w;
  }
  sg[threadIdx.x]=ag; st[threadIdx.x]=at; __syncthreads();
  for(int s=128;s>0;s>>=1){
    if(threadIdx.x<s){ sg[threadIdx.x]+=sg[threadIdx.x+s]; st[threadIdx.x]+=st[threadIdx.x+s]; }
    __syncthreads();
  }
  if(threadIdx.x==0){
    cst[0]=(float)(1.0/sqrt(sg[0]*dz));
    cst[1]=(float)(1.0/sqrt(st[0]*dz));
  }
}

// ------------------------------ SH kernel ----------------------------------

__global__ void sh_kernel(const float* __restrict__ pos, const int* __restrict__ snd,
                          const int* __restrict__ rcv, const float* __restrict__ cg,
                          float* __restrict__ sh, int E){
  int e = blockIdx.x*blockDim.x + threadIdx.x;
  if(e>=E) return;
  int s = snd[e], r = rcv[e];
  float dx = pos[r*3+0]-pos[s*3+0];
  float dy = pos[r*3+1]-pos[s*3+1];
  float dz = pos[r*3+2]-pos[s*3+2];
  float inv = rsqrtf(dx*dx+dy*dy+dz*dz);
  float vx=dx*inv, vy=dy*inv, vz=dz*inv;
  const float s3 = 1.7320508075688772f;
  float y1[3] = { s3*vy, s3*vz, s3*vx };
  float y2[5];
  { const float* C = cg + CG112; float nn=0.f;
    #pragma unroll
    for(int k=0;k<5;k++){ float a=0.f;
      for(int i=0;i<3;i++) for(int j=0;j<3;j++) a += y1[i]*y1[j]*C[(i*3+j)*5+k];
      y2[k]=a; nn+=a*a; }
    float f = 2.2360679774997896f*rsqrtf(nn); // sqrt(5)/||.||
    for(int k=0;k<5;k++) y2[k]*=f; }
  float y3[7];
  { const float* C = cg + CG213; float nn=0.f;
    #pragma unroll
    for(int k=0;k<7;k++){ float a=0.f;
      for(int i=0;i<5;i++) for(int j=0;j<3;j++) a += y2[i]*y1[j]*C[(i*3+j)*7+k];
      y3[k]=a; nn+=a*a; }
    float f = 2.6457513110645906f*rsqrtf(nn); // sqrt(7)/||.||
    for(int k=0;k<7;k++) y3[k]*=f; }
  float* o = sh + (size_t)e*16;
  for(int k=0;k<3;k++) o[k]   = y1[k];
  for(int k=0;k<5;k++) o[3+k] = y2[k];
  for(int k=0;k<7;k++) o[8+k] = y3[k];
  o[15] = 0.f;
}

// ------------------------------ layer 1 ------------------------------------

__global__ void l1_edge_kernel(const float* __restrict__ sh, const int* __restrict__ rcv,
                               float* __restrict__ agg1, int E){
  int e = blockIdx.x*blockDim.x + threadIdx.x;
  if(e>=E) return;
  float* o = agg1 + (size_t)rcv[e]*16;
  atomicAdd(o, 1.0f);
  const float* y = sh + (size_t)e*16;
  #pragma unroll
  for(int i=0;i<15;i++) atomicAdd(o+1+i, y[i]);
}

__device__ inline float geluf(float x){
  return 0.5f*x*(1.0f + tanhf(0.79788456080286535f*(x + 0.044715f*x*x*x)));
}

__global__ void l1_node_kernel(const float* __restrict__ agg1,
    const float* __restrict__ pre0p, const float* __restrict__ pre1m, const float* __restrict__ pre2p,
    const float* __restrict__ post0p, const float* __restrict__ post1m, const float* __restrict__ post2p,
    const float* __restrict__ sc0p, const float* __restrict__ cst,
    float* __restrict__ xout, int N){
  int n = blockIdx.x*blockDim.x + threadIdx.x;
  if(n>=N) return;
  const float rs32 = 0.17677669529663687f; // 1/sqrt(32)
  const float rs8  = 0.35355339059327373f; // 1/sqrt(8)
  float a[16];
  for(int i=0;i<16;i++) a[i] = agg1[(size_t)n*16+i]*DENOM_INV;
  float invCg = cst[0];
  float g[32];
  for(int o=0;o<32;o++) g[o] = geluf(a[0]*pre0p[o])*invCg;
  float* X = xout + (size_t)n*192;
  for(int o=0;o<32;o++){
    float r=0.f;
    for(int u=0;u<32;u++) r += g[u]*post0p[u*32+o];
    X[X0P+o] = r*rs32 + sc0p[o];  // shortcut fan=1
    X[X0M+o] = 0.f;               // (0,-1) structurally zero after layer 1
  }
  for(int o=0;o<8;o++)
    for(int k=0;k<3;k++){
      float r=0.f;
      for(int u=0;u<8;u++) r += (a[1+k]*pre1m[u])*post1m[u*8+o];
      X[X1M+o*3+k] = r*rs8;
      X[X1P+o*3+k] = 0.f;
    }
  for(int o=0;o<8;o++)
    for(int k=0;k<5;k++){
      float r=0.f;
      for(int u=0;u<8;u++) r += (a[4+k]*pre2p[u])*post2p[u*8+o];
      X[X2P+o*5+k] = r*rs8;
      X[X2M+o*5+k] = 0.f;
    }
}

// --------------------- layers 2/3: edge TP + scatter -----------------------

__global__ void edge_tp_kernel(const float* __restrict__ x, const float* __restrict__ sh,
                               const int* __restrict__ snd, const int* __restrict__ rcv,
                               const float* __restrict__ cg, float* __restrict__ agg,
                               int E, int l0only){
  int wv  = (blockIdx.x*blockDim.x + threadIdx.x) >> 5;   // one wave32 per edge
  int lane = threadIdx.x & 31;
  if(wv>=E) return;
  const float* xs = x + (size_t)snd[wv]*192;
  const float* y  = sh + (size_t)wv*16;
  float* out = agg + (size_t)rcv[wv]*1456;

  // passthrough (node channels -> e_ch head of each block)
  int nb = l0only ? 2 : 6;
  for(int b=0;b<nb;b++){
    int so=PASSB[b][0], dofs=PASSB[b][1], len=PASSB[b][2];
    for(int c=lane;c<len;c+=32) atomicAdd(out+dofs+c, xs[so+c]);
  }
  // mul=32 paths: scalar x0 times y (cg(0,l,l)=identity)
  if(!l0only){
    for(int p=0;p<4;p++){
      int so=MUL32P[p][0], yo=MUL32P[p][1], nk=MUL32P[p][2], dofs=MUL32P[p][3];
      int tot = 32*nk;
      for(int idx=lane; idx<tot; idx+=32){
        int u=idx/nk, k=idx-u*nk;
        atomicAdd(out + dofs + (8+u)*nk + k, xs[so+u]*y[yo+k]);
      }
    }
  }
  // mul=8 CG paths
  int np = l0only ? 4 : 26;
  for(int p=0;p<np;p++){
    int so=PATHS[p][0], l1=PATHS[p][1], yo=PATHS[p][2], l2=PATHS[p][3];
    int co=PATHS[p][4], l3=PATHS[p][5], dofs=PATHS[p][6], ub=PATHS[p][7];
    int n1=2*l1+1, n2=2*l2+1, n3=2*l3+1;
    const float* C = cg + co;
    int tot = 8*n3;
    for(int idx=lane; idx<tot; idx+=32){
      int u=idx/n3, k=idx-u*n3;
      float a=0.f;
      for(int i=0;i<n1;i++){
        float xv = xs[so + u*n1 + i];
        for(int j=0;j<n2;j++) a += xv*y[yo+j]*C[(i*n2+j)*n3+k];
      }
      atomicAdd(out + dofs + (ub+u)*n3 + k, a);
    }
  }
}

// --------------------- layer 2 node update (WMMA) --------------------------

struct L2W {
  const float* pre[6];   // [0p,0m,1p,1m,2p,2m]: 48x32,48x32,80x8,80x8,88x8,88x8
  const float* post[6];  // 32x32,32x32,8x8,8x8,8x8,8x8
  const float* sc[6];    // 32x32,32x32,8x8,8x8,8x8,8x8
};

__global__ __launch_bounds__(32)
void l2_node_kernel(const float* __restrict__ agg, const float* __restrict__ xin,
                    L2W w, const float* __restrict__ cst, float* __restrict__ xout, int N){
  __shared__ float hs[16*32];  // activated l0 block, 16 nodes x 32 ch
  const int lane = threadIdx.x;
  const int nb   = blockIdx.x*16;
  const int lc   = lane & 15;          // A-row (node) / B-col (out ch)
  const int kb   = (lane >> 4) * 2;    // K sub-slot per ISA 16x16x4 f32 layout
  const int mofs = 8*(lane >> 4);      // D-row base
  int nrow = nb + lc; if(nrow >= N) nrow = N-1;  // clamp (stores guarded)

  const float invCg = cst[0], invCt = cst[1];
  const float rs48 = 0.14433756729740643f;
  const float rs32 = 0.17677669529663687f;
  const float rs8  = 0.35355339059327373f;
  const float rs80 = 0.11180339887498948f;
  const float rs88 = 0.10660035817780521f;
  const float scalePre0 = rs48*DENOM_INV;

  // ---- l=0 blocks: pre (48->32) -> act -> post (32->32) + shortcut (32->32)
  for(int par=0; par<2; par++){
    const float* A  = agg + (size_t)nrow*1456 + par*48;
    const float* Wp = w.pre[par];
    v8f acc0 = {0,0,0,0,0,0,0,0};
    v8f acc1 = {0,0,0,0,0,0,0,0};
    #pragma unroll
    for(int kk=0; kk<12; kk++){
      v2f a; a.x = A[4*kk+kb]; a.y = A[4*kk+kb+1];
      v2f b0; b0.x = Wp[(4*kk+kb)*32 + lc];      b0.y = Wp[(4*kk+kb+1)*32 + lc];
      v2f b1; b1.x = Wp[(4*kk+kb)*32 + 16 + lc]; b1.y = Wp[(4*kk+kb+1)*32 + 16 + lc];
      acc0 = __builtin_amdgcn_wmma_f32_16x16x4_f32(false, a, false, b0, (short)0, acc0, false, false);
      acc1 = __builtin_amdgcn_wmma_f32_16x16x4_f32(false, a, false, b1, (short)0, acc1, false, false);
    }
    // activation + stage to LDS for the post matmul (D layout -> A layout)
    #pragma unroll
    for(int v=0; v<8; v++){
      float h0 = acc0[v]*scalePre0;
      float h1 = acc1[v]*scalePre0;
      float g0, g1;
      if(par==0){ g0 = geluf(h0)*invCg; g1 = geluf(h1)*invCg; }
      else      { g0 = tanhf(h0)*invCt; g1 = tanhf(h1)*invCt; }
      hs[(mofs+v)*32 + lc]      = g0;
      hs[(mofs+v)*32 + 16 + lc] = g1;
    }
    __syncthreads();
    const float* Wq = w.post[par];
    const float* Ws = w.sc[par];
    const float* Xi = xin + (size_t)nrow*192 + par*32;
    v8f o0 = {0,0,0,0,0,0,0,0};
    v8f o1 = {0,0,0,0,0,0,0,0};
    #pragma unroll
    for(int kk=0; kk<8; kk++){
      v2f a; a.x = hs[lc*32 + 4*kk+kb]; a.y = hs[lc*32 + 4*kk+kb+1];
      v2f b0; b0.x = Wq[(4*kk+kb)*32 + lc];      b0.y = Wq[(4*kk+kb+1)*32 + lc];
      v2f b1; b1.x = Wq[(4*kk+kb)*32 + 16 + lc]; b1.y = Wq[(4*kk+kb+1)*32 + 16 + lc];
      o0 = __builtin_amdgcn_wmma_f32_16x16x4_f32(false, a, false, b0, (short)0, o0, false, false);
      o1 = __builtin_amdgcn_wmma_f32_16x16x4_f32(false, a, false, b1, (short)0, o1, false, false);
    }
    #pragma unroll
    for(int kk=0; kk<8; kk++){
      v2f a; a.x = Xi[4*kk+kb]; a.y = Xi[4*kk+kb+1];
      v2f b0; b0.x = Ws[(4*kk+kb)*32 + lc];      b0.y = Ws[(4*kk+kb+1)*32 + lc];
      v2f b1; b1.x = Ws[(4*kk+kb)*32 + 16 + lc]; b1.y = Ws[(4*kk+kb+1)*32 + 16 + lc];
      o0 = __builtin_amdgcn_wmma_f32_16x16x4_f32(false, a, false, b0, (short)0, o0, false, false);
      o1 = __builtin_amdgcn_wmma_f32_16x16x4_f32(false, a, false, b1, (short)0, o1, false, false);
    }
    #pragma unroll
    for(int v=0; v<8; v++){
      int m = mofs+v, nn = nb+m;
      if(nn < N){
        xout[(size_t)nn*192 + par*32 + lc]      = o0[v]*rs32;  // post & sc both fan 32
        xout[(size_t)nn*192 + par*32 + 16 + lc] = o1[v]*rs32;
      }
    }
    __syncthreads();
  }

  // ---- l=1 / l=2 blocks (VALU; small 8-wide outputs)
  const int aggOff[2][2] = { {96,336}, {576,1016} };
  const int xOff [2][2] = { {X1P,X1M}, {X2P,X2M} };
  bool wr = (lane < 16) && (nb + lc < N);   // lanes 16..31 duplicate work, skip writes
  for(int li=0; li<2; li++){
    int nk  = (li==0)?3:5;
    int fan = (li==0)?80:88;
    float sc = ((li==0)?rs80:rs88)*DENOM_INV;
    for(int par=0; par<2; par++){
      const float* A  = agg + (size_t)nrow*1456 + aggOff[li][par];
      const float* Wp = w.pre [2+2*li+par];
      const float* Wq = w.post[2+2*li+par];
      const float* Ws = w.sc  [2+2*li+par];
      const float* Xi = xin + (size_t)nrow*192 + xOff[li][par];
      float acc[8][5];
      for(int o=0;o<8;o++) for(int k=0;k<5;k++) acc[o][k]=0.f;
      for(int u=0;u<fan;u++){
        const float* wrow = Wp + u*8;
        for(int k=0;k<nk;k++){
          float av = A[u*nk+k];
          for(int o=0;o<8;o++) acc[o][k] += av*wrow[o];
        }
      }
      for(int o=0;o<8;o++)
        for(int k=0;k<nk;k++){
          float r = 0.f;
          for(int u=0;u<8;u++)
            r += acc[u][k]*sc*Wq[u*8+o] + Xi[u*nk+k]*Ws[u*8+o];
          r *= rs8;  // post fan 8 and shortcut fan 8
          if(wr) xout[(size_t)(nb+lc)*192 + xOff[li][par] + o*nk + k] = r;
        }
    }
  }
}

// --------------------- layer 3 node update + graph reduce ------------------

__global__ void l3_node_kernel(const float* __restrict__ agg, const float* __restrict__ xin,
    const float* __restrict__ preM, const float* __restrict__ preP,
    const float* __restrict__ postM, const float* __restrict__ postP,
    const float* __restrict__ scM, const float* __restrict__ scP,
    const float* __restrict__ cst, const int* __restrict__ gid,
    float* __restrict__ pred, int N){
  int n = blockIdx.x*blockDim.x + threadIdx.x;
  if(n>=N) return;
  const float rs48 = 0.14433756729740643f;
  const float rs32 = 0.17677669529663687f;
  const float rs7  = 0.37796447300922722f;
  const float* A = agg + (size_t)n*1456;
  float hm = 0.f;
  for(int u=0;u<48;u++) hm += A[48+u]*preM[u];
  hm *= rs48*DENOM_INV;
  float hp[7] = {0,0,0,0,0,0,0};
  for(int u=0;u<48;u++){
    float av = A[u];
    for(int o=0;o<7;o++) hp[o] += av*preP[u*7+o];
  }
  for(int o=0;o<7;o++) hp[o] *= rs48*DENOM_INV;
  float tm = tanhf(hm)*cst[1];
  float gp[7];
  for(int o=0;o<7;o++) gp[o] = geluf(hp[o])*cst[0];
  float xm = tm*postM[0];  // fan=1
  float xp[7];
  for(int o=0;o<7;o++){
    float r=0.f;
    for(int u=0;u<7;u++) r += gp[u]*postP[u*7+o];
    xp[o] = r*rs7;
  }
  const float* Xi = xin + (size_t)n*192;
  float scm = 0.f;
  for(int u=0;u<32;u++) scm += Xi[X0M+u]*scM[u];
  scm *= rs32;
  float scp[7];
  for(int o=0;o<7;o++){
    float r=0.f;
    for(int u=0;u<32;u++) r += Xi[X0P+u]*scP[u*7+o];
    scp[o] = r*rs32;
  }
  float* pg = pred + (size_t)gid[n]*8;
  atomicAdd(pg+0, xm+scm);
  for(int o=0;o<7;o++) atomicAdd(pg+1+o, xp[o]+scp[o]);
}

__global__ void final_kernel(const float* __restrict__ pred, float* __restrict__ out, int G){
  int g = blockIdx.x*blockDim.x + threadIdx.x;
  if(g>=G) return;
  float odd = pred[g*8+0], e1 = pred[g*8+1];
  out[g*8+0] =  odd*e1;
  out[g*8+1] = -odd*e1;
  for(int i=0;i<6;i++) out[g*8+2+i] = pred[g*8+2+i];
}

// ------------------------------- launcher ----------------------------------

extern "C" void kernel_launch(void* const* d_in, const int* in_sizes, int n_in,
                              void* d_out, int out_size, void* d_ws, size_t ws_size,
                              hipStream_t stream) {
  if(n_in < 39) return;  // expected: positions + 34 weight leaves + 4 trailing
  (void)ws_size;
  const float* pos = (const float*)d_in[0];
  // jax-style param flatten: per layer dict keys sorted (post, pre, shortcut), Nones dropped
  const float* W[35];
  for(int i=1;i<=34;i++) W[i] = (const float*)d_in[i];
  const int* snd = (const int*)d_in[n_in-4];
  const int* rcv = (const int*)d_in[n_in-3];
  const int* gid = (const int*)d_in[n_in-2];
  int N = in_sizes[0]/3;
  int E = in_sizes[n_in-4];
  int G = out_size/8;

  float* ws    = (float*)d_ws;
  float* cg    = ws;                 // 1036 floats
  float* cst   = ws + 2048;          // 2 floats
  float* sh    = ws + 4096;          // E*16
  float* nodeA = sh + (size_t)E*16;  // N*192
  float* nodeB = nodeA + (size_t)N*192;
  float* agg   = nodeB + (size_t)N*192;   // N*1456 (also reused as N*16 for layer 1)
  float* pred  = agg + (size_t)N*1456;    // G*8

  cg_init_kernel<<<4,256,0,stream>>>(cg);
  norm_init_kernel<<<1,256,0,stream>>>(cst);
  sh_kernel<<<(E+255)/256,256,0,stream>>>(pos, snd, rcv, cg, sh, E);

  // ---- layer 1
  hipMemsetAsync(agg, 0, (size_t)N*16*sizeof(float), stream);
  l1_edge_kernel<<<(E+255)/256,256,0,stream>>>(sh, rcv, agg, E);
  l1_node_kernel<<<(N+127)/128,128,0,stream>>>(agg, W[7],W[8],W[9], W[1],W[4],W[5], W[10], cst, nodeA, N);

  // ---- layer 2
  hipMemsetAsync(agg, 0, (size_t)N*1456*sizeof(float), stream);
  edge_tp_kernel<<<(E+3)/4,128,0,stream>>>(nodeA, sh, snd, rcv, cg, agg, E, 0);
  L2W w2;
  for(int i=0;i<6;i++){ w2.post[i]=W[11+i]; w2.pre[i]=W[17+i]; w2.sc[i]=W[23+i]; }
  l2_node_kernel<<<(N+15)/16,32,0,stream>>>(agg, nodeA, w2, cst, nodeB, N);

  // ---- layer 3 (only l=0 edge channels needed)
  hipMemsetAsync(agg, 0, (size_t)N*1456*sizeof(float), stream);
  edge_tp_kernel<<<(E+3)/4,128,0,stream>>>(nodeB, sh, snd, rcv, cg, agg, E, 1);
  hipMemsetAsync(pred, 0, (size_t)G*8*sizeof(float), stream);
  l3_node_kernel<<<(N+127)/128,128,0,stream>>>(agg, nodeB, W[31],W[32], W[29],W[30], W[33],W[34],
                                               cst, gid, pred, N);
  final_kernel<<<(G+63)/64,64,0,stream>>>(pred, (float*)d_out, G);
}